// EdgeNetwork_4690104287616
// MI455X (gfx1250) — compile-verified
//
#include <hip/hip_runtime.h>

#define N_ATOMS  50000
#define N_EDGES  100000
#define ATOM_DIM 64
#define BOND_DIM 16
// W layout: 17 channels (16 bond + 1 bias) x 64 K-rows each, 64 output features.
// LDS float layout: flat = (jq*17 + b)*256 + i*4 + kr,  where j = 4*jq + kr.
//  -> per jq: one base pointer; per (b, tile t): static ds offset b*1024 + t*256 bytes (fits imm16).
#define WK_F4S   17408                // 17*16*256/4 float4s = 278528 bytes of LDS

typedef __attribute__((ext_vector_type(2))) float v2f;
typedef __attribute__((ext_vector_type(8))) float v8f;

__global__ void __launch_bounds__(256) zero_out_kernel(float* __restrict__ out, int n) {
    int i = blockIdx.x * blockDim.x + threadIdx.x;
    if (i < n) out[i] = 0.0f;
}

__global__ void __launch_bounds__(256) edge_net_kernel(
    const float* __restrict__ atom,   // [N_ATOMS, 64]
    const float* __restrict__ bond,   // [N_EDGES, 16]
    const int*   __restrict__ pairs,  // [N_EDGES, 2] (src, dst)
    const float* __restrict__ Kmat,   // [16, 4096]  K[b][i*64+j]
    const float* __restrict__ bias,   // [4096]      bias[i*64+j]
    float* __restrict__ out)          // [N_ATOMS, 64]
{
    __shared__ float4 WK4[WK_F4S];
    const float* WK = (const float*)WK4;

    // Cooperative fill, vectorized b128 both sides:
    // Kmat float4 index f = b*1024 + i*16 + jq  ->  WK4[(jq*17 + b)*64 + i]
    const float4* K4 = (const float4*)Kmat;
    for (int f = threadIdx.x; f < BOND_DIM * 1024; f += blockDim.x) {
        int b = f >> 10; int rem = f & 1023;
        int i = rem >> 4; int jq = rem & 15;
        WK4[(jq * 17 + b) * 64 + i] = K4[f];
    }
    const float4* B4 = (const float4*)bias;
    for (int f = threadIdx.x; f < 1024; f += blockDim.x) {
        int i = f >> 4; int jq = f & 15;
        WK4[(jq * 17 + 16) * 64 + i] = B4[f];   // bias channel b=16
    }
    __syncthreads();

    const int lane = threadIdx.x & 31;
    const int l15  = lane & 15;
    const int hi   = lane >> 4;                       // 0: K 0,1 / M 0..7 ; 1: K 2,3 / M 8..15
    const int wid  = (blockIdx.x * blockDim.x + threadIdx.x) >> 5;
    const int nwv  = (gridDim.x * blockDim.x) >> 5;
    const int npairs = (N_EDGES + 31) / 32;           // 3125 (exact)

    // Lane-dependent part of the B-fragment LDS address (floats): i*4 + kr, i = 16t + l15, kr = 2*hi.
    const int blane = l15 * 4 + hi * 2;

    for (int p = wid; p < npairs; p += nwv) {
        const int gb0 = p * 32;
        const int gb1 = p * 32 + 16;

        // This lane's edge in each 16-edge group (A-matrix row M = l15).
        int e0 = gb0 + l15, e1 = gb1 + l15;
        int ec0 = e0 < N_EDGES ? e0 : N_EDGES - 1;
        int ec1 = e1 < N_EDGES ? e1 : N_EDGES - 1;
        int dst0 = pairs[2 * ec0 + 1];
        int dst1 = pairs[2 * ec1 + 1];
        const float* nb0 = atom + (size_t)dst0 * 64 + hi * 2;
        const float* nb1 = atom + (size_t)dst1 * 64 + hi * 2;

        // Preload this edge's 16 bond weights into registers (b128 loads).
        float bw0[16], bw1[16];
        const float4* bp0 = (const float4*)(bond + (size_t)ec0 * 16);
        const float4* bp1 = (const float4*)(bond + (size_t)ec1 * 16);
        #pragma unroll
        for (int q = 0; q < 4; ++q) {
            float4 t0 = bp0[q], t1 = bp1[q];
            bw0[4*q+0] = t0.x; bw0[4*q+1] = t0.y; bw0[4*q+2] = t0.z; bw0[4*q+3] = t0.w;
            bw1[4*q+0] = t1.x; bw1[4*q+1] = t1.y; bw1[4*q+2] = t1.z; bw1[4*q+3] = t1.w;
        }

        v8f acc0[4], acc1[4];
        #pragma unroll
        for (int t = 0; t < 4; ++t) {
            acc0[t] = (v8f){0,0,0,0,0,0,0,0};
            acc1[t] = (v8f){0,0,0,0,0,0,0,0};
        }

        // K-loop: j-window (4 wide) outer so each neighbor pair is loaded once and
        // reused across all 17 channels; each B frag serves both edge groups (8 WMMA / ds_b64 set).
        const float* wptr = WK + blane;               // + jq*4352 floats per step
        for (int jq = 0; jq < 16; ++jq, wptr += 17 * 256) {
            v2f a0 = *(const v2f*)(nb0 + jq * 4);     // nbr[e, 4jq + hi*2 .. +1]
            v2f a1 = *(const v2f*)(nb1 + jq * 4);

            #pragma unroll
            for (int b = 0; b < 17; ++b) {
                v2f A0, A1;
                if (b < 16) { A0 = a0 * bw0[b]; A1 = a1 * bw1[b]; }
                else        { A0 = a0;          A1 = a1;          }   // bias channel, weight 1
                #pragma unroll
                for (int t = 0; t < 4; ++t) {
                    // static offset b*256 + t*64 floats = b*1024 + t*256 bytes -> ds imm offset
                    v2f B = *(const v2f*)(wptr + b * 256 + t * 64);
                    acc0[t] = __builtin_amdgcn_wmma_f32_16x16x4_f32(
                        false, A0, false, B, (short)0, acc0[t], false, false);
                    acc1[t] = __builtin_amdgcn_wmma_f32_16x16x4_f32(
                        false, A1, false, B, (short)0, acc1[t], false, false);
                }
            }
        }

        // Epilogue: D layout -> lane holds (M = r + hi*8, N = 16*t + l15).
        #pragma unroll
        for (int r = 0; r < 8; ++r) {
            int ea = gb0 + r + hi * 8;
            if (ea < N_EDGES) {
                int s = pairs[2 * ea];
                float* op = out + (size_t)s * 64 + l15;
                unsafeAtomicAdd(op +  0, acc0[0][r]);
                unsafeAtomicAdd(op + 16, acc0[1][r]);
                unsafeAtomicAdd(op + 32, acc0[2][r]);
                unsafeAtomicAdd(op + 48, acc0[3][r]);
            }
            int eb = gb1 + r + hi * 8;
            if (eb < N_EDGES) {
                int s = pairs[2 * eb];
                float* op = out + (size_t)s * 64 + l15;
                unsafeAtomicAdd(op +  0, acc1[0][r]);
                unsafeAtomicAdd(op + 16, acc1[1][r]);
                unsafeAtomicAdd(op + 32, acc1[2][r]);
                unsafeAtomicAdd(op + 48, acc1[3][r]);
            }
        }
    }
}

extern "C" void kernel_launch(void* const* d_in, const int* in_sizes, int n_in,
                              void* d_out, int out_size, void* d_ws, size_t ws_size,
                              hipStream_t stream) {
    const float* atom  = (const float*)d_in[0];
    const float* bondf = (const float*)d_in[1];
    const int*   pairs = (const int*)  d_in[2];
    const float* Kmat  = (const float*)d_in[3];
    const float* bias  = (const float*)d_in[4];
    float* out = (float*)d_out;

    const int outN = N_ATOMS * ATOM_DIM;
    zero_out_kernel<<<(outN + 255) / 256, 256, 0, stream>>>(out, outN);
    // 256 WGs (>= expected WGP count), 8 waves each, grid-stride over 3125 32-edge pairs.
    edge_net_kernel<<<256, 256, 0, stream>>>(atom, bondf, pairs, Kmat, bias, out);
}